// TemporalGNN_50972671869166
// MI455X (gfx1250) — compile-verified
//
#include <hip/hip_runtime.h>
#include <hip/hip_bf16.h>

#define NN 10000
#define EE 160000
#define BB 16
#define FF 2
#define PP 12
#define CC 32
#define FP 24  // F*P channels per (b,n)

typedef __attribute__((ext_vector_type(2))) float v2f;
typedef __attribute__((ext_vector_type(8))) float v8f;

// ---------------- workspace layout (floats) ----------------
#define S_OFF    0                       // B*N*24 = 3,840,000
#define H_OFF    3840000                 // B*N*32 = 5,120,000
#define DEG_OFF  8960000                 // N
#define NORM_OFF 8970000                 // E
#define CST_OFF  9130000                 // Az(64) cz(32) Ah(64) ch(32) probs(12)

__global__ void fill_f32(float* p, float v, int n) {
    int i = blockIdx.x * blockDim.x + threadIdx.x;
    if (i < n) p[i] = v;
}

// Fold gate weights: Az[f][c] = sum_k W_z[f][k]*Lz[k][c]; cz[c] = sum_k b_z[k]*Lz[k][c] + lzb[c]
// (only top half of lin_*_W matters since H0==0). Also softmax(attention).
__global__ void fold_consts(const float* Wz, const float* bz, const float* Lz, const float* lzb,
                            const float* Wh, const float* bh, const float* Lh, const float* lhb,
                            const float* att, float* cst) {
    int t = threadIdx.x;
    if (t < CC) {
        float az0 = 0.f, az1 = 0.f, czv = lzb[t];
        float ah0 = 0.f, ah1 = 0.f, chv = lhb[t];
        for (int k = 0; k < CC; ++k) {
            float lz = Lz[k * CC + t], lh = Lh[k * CC + t];
            az0 += Wz[k] * lz;          // W_z[0][k]
            az1 += Wz[CC + k] * lz;     // W_z[1][k]
            czv += bz[k] * lz;
            ah0 += Wh[k] * lh;
            ah1 += Wh[CC + k] * lh;
            chv += bh[k] * lh;
        }
        cst[t] = az0; cst[32 + t] = az1; cst[64 + t] = czv;
        cst[96 + t] = ah0; cst[128 + t] = ah1; cst[160 + t] = chv;
    }
    if (t < PP) {  // softmax over 12 values, computed redundantly per lane
        float m = -1e30f;
        for (int p = 0; p < PP; ++p) m = fmaxf(m, att[p]);
        float s = 0.f;
        for (int p = 0; p < PP; ++p) s += __expf(att[p] - m);
        cst[192 + t] = __expf(att[t] - m) / s;
    }
}

// deg[dst] += w   (deg pre-filled with 1.0 = self-loop weight)
__global__ void deg_accum(const int* ei, const float* w, float* deg) {
    int e = blockIdx.x * blockDim.x + threadIdx.x;
    if (e < EE) atomicAdd(&deg[ei[EE + e]], w[e]);
}

__global__ void deg_rsqrt(float* deg) {
    int n = blockIdx.x * blockDim.x + threadIdx.x;
    if (n < NN) {
        float d = deg[n];
        deg[n] = (d > 0.f) ? __frsqrt_rn(d) : 0.f;   // now holds dinv
    }
}

__global__ void edge_norm(const int* ei, const float* w, const float* dinv, float* norm) {
    int e = blockIdx.x * blockDim.x + threadIdx.x;
    if (e < EE) norm[e] = dinv[ei[e]] * w[e] * dinv[ei[EE + e]];
}

// Scatter all F*P=24 raw channels:  S[b][dst][c] += norm[e] * x[b][src][c]
__global__ void edge_scatter(const int* ei, const float* norm, const float* x, float* S) {
    long long i = (long long)blockIdx.x * blockDim.x + threadIdx.x;
    const long long total = (long long)EE * BB * FP;
    if (i >= total) return;
    int c = (int)(i % FP);
    long long t = i / FP;
    int b = (int)(t % BB);
    int e = (int)(t / BB);
    int s = ei[e], d = ei[EE + e];
    float v = x[((long long)b * NN + s) * FP + c] * norm[e];
    atomicAdd(&S[((long long)b * NN + d) * FP + c], v);
}

// Self loop: S[b][n][c] += dinv[n]^2 * x[b][n][c]   (runs after scatter; sole writer)
__global__ void self_loop(const float* x, const float* dinv, float* S) {
    long long i = (long long)blockIdx.x * blockDim.x + threadIdx.x;
    const long long total = (long long)BB * NN * FP;
    if (i >= total) return;
    int n = (int)((i / FP) % NN);
    float di = dinv[n];
    S[i] += di * di * x[i];
}

// One wave per (b,n), lane = channel c in [0,32).
// acc = sum_p probs[p] * (1 - sigmoid(s0*Az0+s1*Az1+cz)) * tanh(s0*Ah0+s1*Ah1+ch)
// Hrelu = max(acc, 0)
__global__ void gates(const float* S, const float* cst, float* Hrelu) {
    int tid = blockIdx.x * blockDim.x + threadIdx.x;
    int c = tid & 31;
    int row = tid >> 5;                  // b*N + n
    if (row >= BB * NN) return;
    float az0 = cst[c], az1 = cst[32 + c], czv = cst[64 + c];
    float ah0 = cst[96 + c], ah1 = cst[128 + c], chv = cst[160 + c];
    const float* s = S + (long long)row * FP;
    float acc = 0.f;
#pragma unroll
    for (int p = 0; p < PP; ++p) {
        float s0 = s[p];                 // f=0 (uniform across wave -> broadcast)
        float s1 = s[PP + p];            // f=1
        float zz = fmaf(az0, s0, fmaf(az1, s1, czv));
        float hh = fmaf(ah0, s0, fmaf(ah1, s1, chv));
        float Zv = __frcp_rn(1.f + __expf(-zz));
        float Ht = tanhf(hh);
        acc = fmaf(cst[192 + p], (1.f - Zv) * Ht, acc);
    }
    Hrelu[(long long)row * CC + c] = fmaxf(acc, 0.f);
}

// out[row][p] = Hrelu[row][:] @ Wout[:, p] + bout[p]  via V_WMMA_F32_16X16X4_F32.
// One wave per 16-row tile; K=32 in 8 steps of 4; N=12 padded to 16.
// B matrix + bias staged once per block into LDS (zero-padded) -> branchless,
// conflict-free reads; no divergent exec around the WMMA chain.
// 160000 rows / 16 = 10000 tiles exactly -> EXEC all-ones around WMMA.
__global__ void out_gemm(const float* __restrict__ Hrelu, const float* __restrict__ Wout,
                         const float* __restrict__ bout, float* __restrict__ out) {
    __shared__ float Bs[CC * 16];   // Wout padded to [32][16]
    __shared__ float bs[16];        // bias padded to 16
    int t = threadIdx.x;
    for (int i = t; i < CC * 16; i += 256) {
        int k = i >> 4, n = i & 15;
        Bs[i] = (n < PP) ? Wout[k * PP + n] : 0.f;
    }
    if (t < 16) bs[t] = (t < PP) ? bout[t] : 0.f;
    __syncthreads();

    int lane = t & 31;
    int tile = (blockIdx.x * blockDim.x + t) >> 5;
    int row0 = tile * 16;
    int n = lane & 15;        // output column (N dim), valid for n<12
    int hi = lane >> 4;       // half-wave selector
    float bv = bs[n];
    v8f acc;
#pragma unroll
    for (int v = 0; v < 8; ++v) acc[v] = bv;

    // A fragment: M = lane&15, a.x = A[M, k0+2*hi], a.y = A[M, k0+2*hi+1]
    const float* arow = Hrelu + (long long)(row0 + n) * CC;
#pragma unroll
    for (int k0 = 0; k0 < CC; k0 += 4) {
        int kb = k0 + 2 * hi;
        v2f a, b;
        a.x = arow[kb];
        a.y = arow[kb + 1];
        // B fragment (4x16, K x N): VGPR0 -> K=kb, VGPR1 -> K=kb+1, column n
        b.x = Bs[kb * 16 + n];
        b.y = Bs[(kb + 1) * 16 + n];
        acc = __builtin_amdgcn_wmma_f32_16x16x4_f32(
            /*neg_a=*/false, a, /*neg_b=*/false, b,
            /*c_mod=*/(short)0, acc, /*reuse_a=*/false, /*reuse_b=*/false);
    }
    // C/D layout: element v -> M = v (lanes 0-15) or v+8 (lanes 16-31), column n
    if (n < PP) {
#pragma unroll
        for (int v = 0; v < 8; ++v) {
            int M = hi ? (v + 8) : v;
            out[(long long)(row0 + M) * PP + n] = acc[v];
        }
    }
}

extern "C" void kernel_launch(void* const* d_in, const int* in_sizes, int n_in,
                              void* d_out, int out_size, void* d_ws, size_t ws_size,
                              hipStream_t stream) {
    const float* x    = (const float*)d_in[0];
    const int*   ei   = (const int*)d_in[1];
    const float* ew   = (const float*)d_in[2];
    const float* Wz   = (const float*)d_in[3];
    const float* bz   = (const float*)d_in[4];
    const float* Lz   = (const float*)d_in[5];
    const float* lzb  = (const float*)d_in[6];
    // d_in[7..10] = W_r / b_r / lin_r_W / lin_r_b : dead code since H0 == 0
    const float* Wh   = (const float*)d_in[11];
    const float* bh   = (const float*)d_in[12];
    const float* Lh   = (const float*)d_in[13];
    const float* lhb  = (const float*)d_in[14];
    const float* att  = (const float*)d_in[15];
    const float* Wout = (const float*)d_in[16];
    const float* bout = (const float*)d_in[17];
    float* out = (float*)d_out;

    float* ws   = (float*)d_ws;
    float* S    = ws + S_OFF;
    float* Hrl  = ws + H_OFF;
    float* deg  = ws + DEG_OFF;   // becomes dinv in-place
    float* norm = ws + NORM_OFF;
    float* cst  = ws + CST_OFF;

    const int T = 256;
    const long long nS = (long long)BB * NN * FP;           // 3,840,000
    const long long nScatter = (long long)EE * BB * FP;     // 61,440,000

    fill_f32<<<(int)((nS + T - 1) / T), T, 0, stream>>>(S, 0.f, (int)nS);
    fill_f32<<<(NN + T - 1) / T, T, 0, stream>>>(deg, 1.0f, NN);   // self-loop weight
    fold_consts<<<1, 64, 0, stream>>>(Wz, bz, Lz, lzb, Wh, bh, Lh, lhb, att, cst);
    deg_accum<<<(EE + T - 1) / T, T, 0, stream>>>(ei, ew, deg);
    deg_rsqrt<<<(NN + T - 1) / T, T, 0, stream>>>(deg);
    edge_norm<<<(EE + T - 1) / T, T, 0, stream>>>(ei, ew, deg, norm);
    edge_scatter<<<(int)((nScatter + T - 1) / T), T, 0, stream>>>(ei, norm, x, S);
    self_loop<<<(int)((nS + T - 1) / T), T, 0, stream>>>(x, deg, S);
    gates<<<(BB * NN * 32) / T, T, 0, stream>>>(S, cst, Hrl);
    out_gemm<<<(BB * NN / 16) * 32 / T, T, 0, stream>>>(Hrl, Wout, bout, out);
}